// BatchTreeEncoder_89421219103663
// MI455X (gfx1250) — compile-verified
//
#include <hip/hip_runtime.h>
#include <hip/hip_bf16.h>
#include <math.h>

typedef float v2f __attribute__((ext_vector_type(2)));
typedef float v8f __attribute__((ext_vector_type(8)));

#define ENC  512
#define MAXD 8
#define MAXS 40

// ---------------------------------------------------------------------------
// Phase 1: dot_w[v] = emb_table[v,:] . w   via V_WMMA_F32_16X16X4_F32
// One wave (32 lanes) per 16 vocab rows. A = 16x4 tile of emb rows,
// B = 4x16 with w in column N=0 only (lane-select done with a 0/1 multiply,
// NOT a divergent branch, so EXEC stays all-ones around the WMMAs).
// D column 0 lives in lane 0 (rows 0..7) and lane 16 (rows 8..15).
// ---------------------------------------------------------------------------
__global__ __launch_bounds__(32) void dotw_wmma_kernel(
    const float* __restrict__ emb, const float* __restrict__ w,
    float* __restrict__ dotw, int V) {
  __shared__ float2 sw2[ENC / 2];
  const int lane = threadIdx.x;
  for (int i = lane; i < ENC / 2; i += 32)
    sw2[i] = reinterpret_cast<const float2*>(w)[i];
  __syncthreads();

  const int m  = lane & 15;   // row within 16-row tile (also N for B/D)
  const int kh = lane >> 4;   // K half: 0 -> K{0,1}, 1 -> K{2,3}
  const int r0 = blockIdx.x * 16;
  int row = r0 + m;
  if (row >= V) row = V - 1;
  const float* rp = emb + (size_t)row * ENC;

  const float msel = (m == 0) ? 1.0f : 0.0f;  // keep only column N=0 of B

  v8f acc = {};
  for (int k = 0; k < ENC; k += 4) {
    const float2 av = *reinterpret_cast<const float2*>(rp + k + 2 * kh);
    const float2 wv = sw2[(k >> 1) + kh];     // one ds_load_b64, all lanes
    v2f a; a.x = av.x;        a.y = av.y;
    v2f b; b.x = msel * wv.x; b.y = msel * wv.y;
    acc = __builtin_amdgcn_wmma_f32_16x16x4_f32(
        /*neg_a=*/false, a, /*neg_b=*/false, b,
        /*c_mod=*/(short)0, acc, /*reuse_a=*/false, /*reuse_b=*/false);
  }

  if (m == 0) {
    const int base = r0 + 8 * kh;
#pragma unroll
    for (int j = 0; j < 8; ++j) {
      const int r = base + j;
      if (r < V) dotw[r] = acc[j];
    }
  }
}

// ---------------------------------------------------------------------------
// Phase 2: bottom-up tree aggregation. One block per tree, 128 threads,
// each thread owns 4 consecutive E-columns (float4 / global_load_b128).
// 128 lanes x 16B = one full 2KB embedding row per gather - fully coalesced
// reads from the L2-resident embedding table (the true bottleneck).
// ---------------------------------------------------------------------------
__global__ __launch_bounds__(128) void tree_encode_kernel(
    const int* __restrict__ tokens, const unsigned char* __restrict__ masks,
    const float* __restrict__ emb, const float* __restrict__ dotw,
    float* __restrict__ out) {
  const int n   = blockIdx.x;
  const int tid = threadIdx.x;

  __shared__ int   s_tok[MAXS];
  __shared__ float s_dotE[MAXS];
  __shared__ float s_attn[MAXS];
  __shared__ float s_logit[MAXS];
  __shared__ float s_gc[MAXS];
  __shared__ int   s_mask[2][MAXS];
  __shared__ int   s_valid[2][MAXS];
  __shared__ float s_g, s_dotA;

  float4 A = make_float4(0.f, 0.f, 0.f, 0.f);  // running aggregate A_prev

  for (int d = MAXD - 1; d >= 0; --d) {
    const int cur = d & 1, prv = cur ^ 1;

    if (tid < MAXS) {
      const size_t base = ((size_t)n * MAXD + d) * MAXS + tid;
      int tok = tokens[base];
      const int mk = masks[base] ? 1 : 0;
      const int vd = (mk && tok >= 0) ? 1 : 0;
      if (tok < 0) tok = 0;                 // clip(tokens, 0)
      s_tok[tid]        = tok;
      s_mask[cur][tid]  = mk;
      s_valid[cur][tid] = vd;
      s_dotE[tid]       = dotw[tok];
    }
    __syncthreads();

    if (tid == 0) {
      if (d == MAXD - 1) {                  // leaf level: masked plain sum
        float dA = 0.f;
        for (int s = 0; s < MAXS; ++s) {
          const float a = s_valid[cur][s] ? 1.f : 0.f;
          s_attn[s] = a;
          dA += a * s_dotE[s];
        }
        s_g = 0.f; s_dotA = dA;
      } else {
        int nc = 0;
        for (int s = 0; s < MAXS; ++s) nc += s_mask[prv][s];
        const float dotA = s_dotA;
        float mx = -3.4e38f;
        for (int s = 0; s < MAXS; ++s) {
          const float de = s_dotE[s];
          const float ga = 1.f / (1.f + expf(-de));            // sigmoid
          const int   c  = (s < nc) && s_valid[cur][s] && s_valid[prv][s];
          const float lo = de + (c ? ga * dotA : 0.f);          // h_d . w
          s_logit[s] = lo;
          s_gc[s]    = c ? ga : 0.f;
          const float ml = s_valid[cur][s] ? lo : -1.0e9f;
          s_attn[s] = ml;                                       // temp storage
          mx = fmaxf(mx, ml);
        }
        float sum = 0.f;
        for (int s = 0; s < MAXS; ++s) {
          const float e = expf(s_attn[s] - mx);
          s_attn[s] = e;
          sum += e;
        }
        const float inv = 1.f / sum;
        float g = 0.f, dA = 0.f;
        for (int s = 0; s < MAXS; ++s) {
          const float a = s_attn[s] * inv;
          s_attn[s] = a;
          g  += a * s_gc[s];        // coefficient of A_prev in new aggregate
          dA += a * s_logit[s];     // new A . w, computed with scalars only
        }
        s_g = g; s_dotA = dA;
      }
    }
    __syncthreads();

    // E-dimension work: A = sum_s attn[s]*emb[tok[s]] + g*A_prev
    float4 acc = make_float4(0.f, 0.f, 0.f, 0.f);
    for (int s = 0; s < MAXS; ++s) {
      const float a = s_attn[s];
      if (((tid & 7) == 0) && (s + 1 < MAXS))   // one lane per 128B line
        __builtin_prefetch(emb + (size_t)s_tok[s + 1] * ENC + 4 * tid, 0, 0);
      if (a != 0.f) {                            // uniform-in-practice branch
        const float4 v = *reinterpret_cast<const float4*>(
            emb + (size_t)s_tok[s] * ENC + 4 * tid);
        acc.x += a * v.x;
        acc.y += a * v.y;
        acc.z += a * v.z;
        acc.w += a * v.w;
      }
    }
    const float gg = s_g;
    A.x = acc.x + gg * A.x;
    A.y = acc.y + gg * A.y;
    A.z = acc.z + gg * A.z;
    A.w = acc.w + gg * A.w;
    __syncthreads();   // protect shared arrays before next level overwrites
  }

  // masks[:,0,0] forced True -> roots = final aggregate
  reinterpret_cast<float4*>(out + (size_t)n * ENC)[tid] = A;
}

extern "C" void kernel_launch(void* const* d_in, const int* in_sizes, int n_in,
                              void* d_out, int out_size, void* d_ws, size_t ws_size,
                              hipStream_t stream) {
  const int*           tokens = (const int*)d_in[0];
  const unsigned char* masks  = (const unsigned char*)d_in[1];
  const float*         emb    = (const float*)d_in[2];
  const float*         w      = (const float*)d_in[3];
  float*               out    = (float*)d_out;

  const int N = in_sizes[0] / (MAXD * MAXS);
  const int V = in_sizes[2] / ENC;

  float* dotw = (float*)d_ws;   // V floats of scratch (40 KB)

  dotw_wmma_kernel<<<dim3((V + 15) / 16), dim3(32), 0, stream>>>(emb, w, dotw, V);
  tree_encode_kernel<<<dim3(N), dim3(128), 0, stream>>>(tokens, masks, emb, dotw, out);
}